// GNNStack_1838246002728
// MI455X (gfx1250) — compile-verified
//
#include <hip/hip_runtime.h>
#include <cstdint>
#include <cstddef>

typedef __attribute__((ext_vector_type(2))) float v2f;
typedef __attribute__((ext_vector_type(8))) float v8f;

#define FD 256   // feature dim (compile-time per reference)

// ---------------- degree / norm ----------------------------------------

__global__ void fill1_kernel(float* __restrict__ p, int n) {
  int i = blockIdx.x * blockDim.x + threadIdx.x;
  if (i < n) p[i] = 1.0f;   // self-loop contributes 1 to in-degree
}

__global__ void degree_kernel(const long long* __restrict__ dst,
                              float* __restrict__ deg, int e) {
  int i = blockIdx.x * blockDim.x + threadIdx.x;
  if (i < e) unsafeAtomicAdd(&deg[dst[i]], 1.0f);
}

__global__ void rsqrt_kernel(float* __restrict__ p, int n) {
  int i = blockIdx.x * blockDim.x + threadIdx.x;
  if (i < n) p[i] = rsqrtf(p[i]);    // deg >= 1 always
}

// ---------------- WMMA GEMM: Y[N,256] = X[N,256] @ W[256,256] ----------
// Block = 256 threads = 8 waves. Each wave computes one 16x16 tile of Y.
// blockIdx.y selects the 16-col slab of W, staged once into LDS.

__global__ void gemm_wmma_kernel(const float* __restrict__ X,
                                 const float* __restrict__ W,
                                 float* __restrict__ Y, int mtiles) {
  __shared__ float Wt[FD * 16];           // 16 KB: W[k][n0+n] -> Wt[k*16+n]
  const int n0 = blockIdx.y * 16;
  for (int i = threadIdx.x; i < FD * 16; i += 256) {
    int k = i >> 4, n = i & 15;
    Wt[i] = W[k * FD + n0 + n];
  }
  __syncthreads();

  const int wave = threadIdx.x >> 5;
  const int lane = threadIdx.x & 31;
  const int mtile = blockIdx.x * 8 + wave;
  if (mtile >= mtiles) return;            // wave-uniform exit: EXEC stays all-1s
  const int m0 = mtile * 16;

  const int half = lane >> 4;             // lanes 16-31 hold K+2/K+3
  const int l    = lane & 15;
  const float* arow = X + (size_t)(m0 + l) * FD;

  v8f c = {};
  for (int k = 0; k < FD; k += 4) {
    const int ka = k + 2 * half;
    v2f a = *(const v2f*)(arow + ka);                       // A: 16x4 fp32
    v2f b;                                                  // B: 4x16 fp32
    b.x = Wt[ka * 16 + l];
    b.y = Wt[(ka + 1) * 16 + l];
    c = __builtin_amdgcn_wmma_f32_16x16x4_f32(
        /*neg_a=*/false, a, /*neg_b=*/false, b,
        /*c_mod=*/(short)0, c, /*reuse_a=*/false, /*reuse_b=*/false);
  }

  // C/D layout: VGPR v, lanes 0-15 -> M=v, lanes 16-31 -> M=v+8; N=l
  float* yrow = Y + (size_t)m0 * FD + n0 + l;
#pragma unroll
  for (int vg = 0; vg < 8; ++vg)
    yrow[(size_t)(vg + 8 * half) * FD] = c[vg];
}

// ---------------- edge gather * norm -> scatter-add --------------------
// One wave per edge; each lane moves 8 contiguous floats.

__global__ void scatter_kernel(const float* __restrict__ xw,
                               const long long* __restrict__ src,
                               const long long* __restrict__ dst,
                               const float* __restrict__ dis,
                               float* __restrict__ out, int e) {
  const int gid  = blockIdx.x * blockDim.x + threadIdx.x;
  const int edge = gid >> 5;
  const int lane = gid & 31;
  if (edge >= e) return;
  const long long s = src[edge];
  const long long d = dst[edge];
  const float nrm = dis[s] * dis[d];

  const float4* xp = (const float4*)(xw + (size_t)s * FD + lane * 8);
  float4 a = xp[0], b = xp[1];
  float* op = out + (size_t)d * FD + lane * 8;
  unsafeAtomicAdd(op + 0, a.x * nrm);
  unsafeAtomicAdd(op + 1, a.y * nrm);
  unsafeAtomicAdd(op + 2, a.z * nrm);
  unsafeAtomicAdd(op + 3, a.w * nrm);
  unsafeAtomicAdd(op + 4, b.x * nrm);
  unsafeAtomicAdd(op + 5, b.y * nrm);
  unsafeAtomicAdd(op + 6, b.z * nrm);
  unsafeAtomicAdd(op + 7, b.w * nrm);
}

// ---------------- self-loop + bias + ELU -------------------------------

__global__ void epilogue_kernel(const float* __restrict__ agg,
                                const float* __restrict__ xw,
                                const float* __restrict__ dis,
                                const float* __restrict__ bias,
                                float* __restrict__ out, int total) {
  int idx = blockIdx.x * blockDim.x + threadIdx.x;
  if (idx >= total) return;
  const int i = idx >> 8;        // node
  const int d = idx & 255;       // feature
  const float s = dis[i];
  const float v = agg[idx] + xw[idx] * s * s + bias[d];
  out[idx] = v > 0.0f ? v : expm1f(v);   // ELU(alpha=1)
}

// ---------------- launcher ---------------------------------------------

extern "C" void kernel_launch(void* const* d_in, const int* in_sizes, int n_in,
                              void* d_out, int out_size, void* d_ws, size_t ws_size,
                              hipStream_t stream) {
  const float*     x   = (const float*)d_in[0];
  const long long* ei  = (const long long*)d_in[1];   // int64 per reference
  const float*     W1  = (const float*)d_in[2];
  const float*     b1  = (const float*)d_in[3];
  const float*     W2  = (const float*)d_in[4];
  const float*     b2  = (const float*)d_in[5];
  float*           out = (float*)d_out;

  const int N = in_sizes[0] / FD;        // 100000
  const int E = in_sizes[1] / 2;         // 3200000
  const long long* srcp = ei;            // edge_index[0]
  const long long* dstp = ei + E;        // edge_index[1]

  // workspace: dis[N] | buf0[N*FD] | buf1[N*FD]
  char* ws = (char*)d_ws;
  float* dis = (float*)ws;
  size_t off = ((size_t)N * sizeof(float) + 255) & ~(size_t)255;
  float* buf0 = (float*)(ws + off);
  float* buf1 = buf0 + (size_t)N * FD;

  const size_t featBytes = (size_t)N * FD * sizeof(float);
  const int mtiles = N / 16;                         // N multiple of 16
  const dim3 ggrid((mtiles + 7) / 8, FD / 16);
  const int sgrid = (int)(((long long)E * 32 + 255) / 256);
  const int total = N * FD;

  // --- normalization coefficients ---
  fill1_kernel<<<(N + 255) / 256, 256, 0, stream>>>(dis, N);
  degree_kernel<<<(E + 255) / 256, 256, 0, stream>>>(dstp, dis, E);
  rsqrt_kernel<<<(N + 255) / 256, 256, 0, stream>>>(dis, N);

  // --- layer 1 ---
  gemm_wmma_kernel<<<ggrid, 256, 0, stream>>>(x, W1, buf0, mtiles);
  hipMemsetAsync(buf1, 0, featBytes, stream);
  scatter_kernel<<<sgrid, 256, 0, stream>>>(buf0, srcp, dstp, dis, buf1, E);
  epilogue_kernel<<<(total + 255) / 256, 256, 0, stream>>>(buf1, buf0, dis, b1, buf1, total);

  // --- layer 2 ---
  gemm_wmma_kernel<<<ggrid, 256, 0, stream>>>(buf1, W2, buf0, mtiles);
  hipMemsetAsync(out, 0, featBytes, stream);
  scatter_kernel<<<sgrid, 256, 0, stream>>>(buf0, srcp, dstp, dis, out, E);
  epilogue_kernel<<<(total + 255) / 256, 256, 0, stream>>>(out, buf0, dis, b2, out, total);
}